// LinearKAN_69114613730528
// MI455X (gfx1250) — compile-verified
//
#include <hip/hip_runtime.h>

// Problem constants (match reference)
#define BATCH 2048
#define IN    512
#define OUT   512
#define G     8
#define KK    (2*IN)      // packed GEMM depth: [S | silu(x)] vs [w_s*sum_a(c) | w_b]

typedef __attribute__((ext_vector_type(2))) float v2f;
typedef __attribute__((ext_vector_type(8))) float v8f;

// ---------------------------------------------------------------------------
// Prep 1: per-element activation pack  A[b][0..511]=S, A[b][512..1023]=silu(x)
// ---------------------------------------------------------------------------
__global__ __launch_bounds__(256)
void kan_prep_act(const float* __restrict__ x,
                  const float* __restrict__ grid,
                  float* __restrict__ A) {
  int idx = blockIdx.x * blockDim.x + threadIdx.x;   // flat over b*IN + i
  if (idx >= BATCH * IN) return;
  float xv = x[idx];
  float s = 0.f;
#pragma unroll
  for (int n = 0; n < G; ++n) {
    float d = xv - grid[n];        // grid is uniform -> scalar loads
    s += expf(-d * d);
  }
  float u = xv / (1.f + expf(-xv));  // silu
  int b = idx / IN, i = idx - b * IN;
  A[(size_t)b * KK + i]      = s;
  A[(size_t)b * KK + IN + i] = u;
}

// ---------------------------------------------------------------------------
// Prep 2: weight pack  W[o][0..511]=w_s*sum_a c[a,o,:],  W[o][512..1023]=w_b
// ---------------------------------------------------------------------------
__global__ __launch_bounds__(256)
void kan_prep_w(const float* __restrict__ c,
                const float* __restrict__ w_b,
                const float* __restrict__ w_s,
                float* __restrict__ W) {
  int idx = blockIdx.x * blockDim.x + threadIdx.x;   // flat over o*IN + i
  if (idx >= OUT * IN) return;
  float cs = 0.f;
#pragma unroll
  for (int a = 0; a < G; ++a) cs += c[(size_t)a * OUT * IN + idx];
  int o = idx / IN, i = idx - o * IN;
  W[(size_t)o * KK + i]      = cs * w_s[idx];
  W[(size_t)o * KK + IN + i] = w_b[idx];
}

// ---------------------------------------------------------------------------
// GEMM: out[2048x512] = A[2048x1024] * W[512x1024]^T via V_WMMA_F32_16X16X4_F32
//   block = 256 threads (8 waves), tile 128(M) x 64(N), K-slab 32
//   wave grid 4(M) x 2(N); each wave owns 2x2 accumulators of 16x16
//   double-buffered LDS + register-staged prefetch: ONE barrier per K-slab,
//   next-slab global loads overlap the 32 WMMAs of the current slab.
// ---------------------------------------------------------------------------
#define BM 128
#define BN 64
#define BK 32
#define LDP 36   // padded LDS row stride (floats): bank-conflict-free, 16B aligned
#define SA_SZ (BM * LDP)
#define SB_SZ (BN * LDP)

__global__ __launch_bounds__(256)
void kan_gemm(const float* __restrict__ A,   // [BATCH][KK]
              const float* __restrict__ W,   // [OUT][KK]
              float* __restrict__ Cout) {    // [BATCH][OUT]
  __shared__ float sA[2][SA_SZ];
  __shared__ float sB[2][SB_SZ];

  const int mBase = blockIdx.x * BM;
  const int nBase = blockIdx.y * BN;
  const int t    = threadIdx.x;
  const int lane = t & 31;
  const int wave = t >> 5;                 // 0..7
  const int wm   = (wave & 3) * 32;        // wave M offset in block tile
  const int wn   = (wave >> 2) * 32;       // wave N offset in block tile
  const int hi   = lane >> 4;              // 0: K pair {0,1}, 1: K pair {2,3}
  const int l16  = lane & 15;

  v8f acc[2][2] = {};

  // global->LDS staging coordinates (coalesced float4 runs within a row)
  const int arow = t >> 1;                 // 0..127, 2 threads/row, 16 floats each
  const int acol = (t & 1) * 16;
  const int brow = t >> 2;                 // 0..63, 4 threads/row, 8 floats each
  const int bcol = (t & 3) * 8;

  const float* gA = A + (size_t)(mBase + arow) * KK + acol;
  const float* gW = W + (size_t)(nBase + brow) * KK + bcol;

  // ---- pipeline prologue: stage slab 0 into registers ----
  float4 ra[4], rb[2];
#pragma unroll
  for (int q = 0; q < 4; ++q) ra[q] = *(const float4*)(gA + q * 4);
  rb[0] = *(const float4*)(gW);
  rb[1] = *(const float4*)(gW + 4);

  int p = 0;
  for (int k0 = 0; k0 < KK; k0 += BK) {
    // commit staged registers into LDS buffer p
#pragma unroll
    for (int q = 0; q < 4; ++q)
      *(float4*)(&sA[p][arow * LDP + acol + q * 4]) = ra[q];
    *(float4*)(&sB[p][brow * LDP + bcol])     = rb[0];
    *(float4*)(&sB[p][brow * LDP + bcol + 4]) = rb[1];
    __syncthreads();   // buffer p ready; also fences reuse of buffer p^1

    // issue next slab's global loads NOW -> overlap with WMMA below
    if (k0 + BK < KK) {
      const float* nA = gA + k0 + BK;
      const float* nW = gW + k0 + BK;
#pragma unroll
      for (int q = 0; q < 4; ++q) ra[q] = *(const float4*)(nA + q * 4);
      rb[0] = *(const float4*)(nW);
      rb[1] = *(const float4*)(nW + 4);
      if (k0 + 2 * BK < KK) {
        __builtin_prefetch(gA + k0 + 2 * BK, 0, 1);   // global_prefetch_b8
        __builtin_prefetch(gW + k0 + 2 * BK, 0, 1);
      }
    }

    // 32 WMMAs from LDS buffer p
#pragma unroll
    for (int kk = 0; kk < BK; kk += 4) {
      const int kb = kk + hi * 2;
      v2f af[2], bf[2];
#pragma unroll
      for (int i = 0; i < 2; ++i)   // A fragments: 16x4, M=l16, K={kb,kb+1}
        af[i] = *(const v2f*)(&sA[p][(wm + i * 16 + l16) * LDP + kb]);
#pragma unroll
      for (int j = 0; j < 2; ++j)   // B fragments: 4x16, N=l16 (B[k][n]=W[n][k])
        bf[j] = *(const v2f*)(&sB[p][(wn + j * 16 + l16) * LDP + kb]);
#pragma unroll
      for (int i = 0; i < 2; ++i)
#pragma unroll
        for (int j = 0; j < 2; ++j)
          acc[i][j] = __builtin_amdgcn_wmma_f32_16x16x4_f32(
              false, af[i], false, bf[j], (short)0, acc[i][j], false, false);
    }
    p ^= 1;
  }

  // Epilogue. C/D layout: VGPR r -> lanes 0-15: M=r, lanes 16-31: M=r+8; N=l16
#pragma unroll
  for (int i = 0; i < 2; ++i) {
#pragma unroll
    for (int j = 0; j < 2; ++j) {
      const int col = nBase + wn + j * 16 + l16;
#pragma unroll
      for (int r = 0; r < 8; ++r) {
        const int row = mBase + wm + i * 16 + r + hi * 8;
        Cout[(size_t)row * OUT + col] = acc[i][j][r];
      }
    }
  }
}

// ---------------------------------------------------------------------------
extern "C" void kernel_launch(void* const* d_in, const int* in_sizes, int n_in,
                              void* d_out, int out_size, void* d_ws, size_t ws_size,
                              hipStream_t stream) {
  const float* x    = (const float*)d_in[0];   // [2048,512]
  const float* c    = (const float*)d_in[1];   // [8,512,512]
  const float* w_b  = (const float*)d_in[2];   // [512,512]
  const float* w_s  = (const float*)d_in[3];   // [512,512]
  const float* grid = (const float*)d_in[4];   // [8]
  float* out = (float*)d_out;                  // [2048,512]

  float* Apack = (float*)d_ws;                          // 2048*1024 f32 = 8 MB
  float* Wpack = Apack + (size_t)BATCH * KK;            //  512*1024 f32 = 2 MB

  kan_prep_act<<<(BATCH * IN + 255) / 256, 256, 0, stream>>>(x, grid, Apack);
  kan_prep_w  <<<(OUT * IN + 255) / 256, 256, 0, stream>>>(c, w_b, w_s, Wpack);

  dim3 g(BATCH / BM, OUT / BN);   // 16 x 8 = 128 workgroups
  kan_gemm<<<g, 256, 0, stream>>>(Apack, Wpack, out);
}